// IntraModalLoss_29858612642236
// MI455X (gfx1250) — compile-verified
//
#include <hip/hip_runtime.h>
#include <hip/hip_bf16.h>
#include <math.h>

typedef __attribute__((ext_vector_type(2))) float v2f;
typedef __attribute__((ext_vector_type(8))) float v8f;

#define NROWS 8192
#define DDIM  256
#define TEMP  0.2f
#define TILE  128
#define KSTAGE 32
#define LDS_STRIDE 36   // 32 data + 4 pad floats: rows 16B-aligned, conflict-free b64 frag reads

// ---------------------------------------------------------------------------
// Kernel 0: zero the scalar accumulators (ws[0]=positives, ws[1]=negatives)
// ---------------------------------------------------------------------------
__global__ void zero_accum_kernel(float* ws) {
    if (threadIdx.x < 2) ws[threadIdx.x] = 0.0f;
}

// ---------------------------------------------------------------------------
// Kernel 1: per-row ||emb_i||, store inv_norm, accumulate positives term.
// One wave per row: 32 lanes x 8 floats = 256 = D.
// ---------------------------------------------------------------------------
__global__ __launch_bounds__(256)
void rownorm_pos_kernel(const float* __restrict__ emb,
                        float* __restrict__ inv_norm,
                        float* __restrict__ pos_accum) {
    const int wave = threadIdx.x >> 5;
    const int lane = threadIdx.x & 31;
    const int row  = blockIdx.x * 8 + wave;

    const float4* rp = (const float4*)(emb + (size_t)row * DDIM);
    float4 a = rp[lane];        // floats [lane*4 .. lane*4+3]
    float4 b = rp[lane + 32];   // floats [128+lane*4 ..]
    float ss = a.x*a.x + a.y*a.y + a.z*a.z + a.w*a.w
             + b.x*b.x + b.y*b.y + b.z*b.z + b.w*b.w;
    #pragma unroll
    for (int m = 16; m >= 1; m >>= 1) ss += __shfl_xor(ss, m, 32);

    if (lane == 0) {
        float nrm = fmaxf(sqrtf(ss), 1e-12f);
        float inv = 1.0f / nrm;
        inv_norm[row] = inv;
        float pos_sim = ss * inv * inv;                 // == sim_ii (1.0, or 0 for zero row)
        atomicAdd(pos_accum, log1pf(expf(-pos_sim + TEMP)));
    }
}

// ---------------------------------------------------------------------------
// Kernel 2: upper-triangular 128x128 tiles of sim = E E^T via f32 WMMA,
// fused softplus epilogue reduced to a scalar (off-diagonal tiles weighted x2).
// 256 threads = 8 waves; wave w computes rows [w*16, w*16+16) x 128 cols
// as 8 accumulators of V_WMMA_F32_16X16X4_F32.
// ---------------------------------------------------------------------------
__global__ __launch_bounds__(256)
void sim_tile_kernel(const float* __restrict__ emb,
                     const float* __restrict__ inv_norm,
                     float* __restrict__ neg_accum) {
    const int bi = blockIdx.y;   // row tile
    const int bj = blockIdx.x;   // col tile
    if (bj < bi) return;         // symmetry: upper triangle only (block-uniform exit)

    __shared__ float ldsA[2][TILE * LDS_STRIDE];
    __shared__ float ldsB[2][TILE * LDS_STRIDE];
    __shared__ float red[256];

    const int t    = threadIdx.x;
    const int wave = t >> 5;
    const int lane = t & 31;
    const int half = lane >> 4;   // K-pair select for A/B frags
    const int lr   = lane & 15;   // M (A) / N (B,C)

    const int rowBase = bi * TILE;
    const int colBase = bj * TILE;

    v8f acc[8];
    #pragma unroll
    for (int n = 0; n < 8; ++n) {
        #pragma unroll
        for (int e = 0; e < 8; ++e) acc[n][e] = 0.0f;
    }

    float4 ra[4], rb[4];

    // ---- load stage 0 (global -> regs -> LDS) ----
    #pragma unroll
    for (int it = 0; it < 4; ++it) {
        int q  = t + it * 256;
        int r  = q >> 3;
        int kq = (q & 7) << 2;
        ra[it] = *(const float4*)(emb + (size_t)(rowBase + r) * DDIM + kq);
        rb[it] = *(const float4*)(emb + (size_t)(colBase + r) * DDIM + kq);
    }
    #pragma unroll
    for (int it = 0; it < 4; ++it) {
        int q = t + it * 256; int r = q >> 3; int kq = (q & 7) << 2;
        *(float4*)&ldsA[0][r * LDS_STRIDE + kq] = ra[it];
        *(float4*)&ldsB[0][r * LDS_STRIDE + kq] = rb[it];
    }
    __syncthreads();

    const int NSTAGE = DDIM / KSTAGE;   // 8
    for (int s = 0; s < NSTAGE; ++s) {
        const int buf = s & 1;

        // prefetch next K-stage into registers (overlaps with WMMA below)
        if (s + 1 < NSTAGE) {
            const int ks = (s + 1) * KSTAGE;
            #pragma unroll
            for (int it = 0; it < 4; ++it) {
                int q = t + it * 256; int r = q >> 3; int kq = (q & 7) << 2;
                ra[it] = *(const float4*)(emb + (size_t)(rowBase + r) * DDIM + ks + kq);
                rb[it] = *(const float4*)(emb + (size_t)(colBase + r) * DDIM + ks + kq);
            }
        }

        // ---- WMMA over this K-stage ----
        const float* A = &ldsA[buf][(wave * 16) * LDS_STRIDE];
        const float* B = &ldsB[buf][0];
        #pragma unroll
        for (int kk = 0; kk < KSTAGE; kk += 4) {
            const int kof = kk + half * 2;
            v2f a = *(const v2f*)(A + lr * LDS_STRIDE + kof);
            #pragma unroll
            for (int n = 0; n < 8; ++n) {
                v2f b = *(const v2f*)(B + (n * 16 + lr) * LDS_STRIDE + kof);
                acc[n] = __builtin_amdgcn_wmma_f32_16x16x4_f32(
                    false, a, false, b, (short)0, acc[n], false, false);
            }
        }

        if (s + 1 < NSTAGE) {
            __syncthreads();   // everyone done reading buf^1 (read in the previous stage)
            const int nb = (s + 1) & 1;
            #pragma unroll
            for (int it = 0; it < 4; ++it) {
                int q = t + it * 256; int r = q >> 3; int kq = (q & 7) << 2;
                *(float4*)&ldsA[nb][r * LDS_STRIDE + kq] = ra[it];
                *(float4*)&ldsB[nb][r * LDS_STRIDE + kq] = rb[it];
            }
            __syncthreads();   // stores visible before next stage's compute
        }
    }

    // ---- fused epilogue: sim -> log1p(exp(sim - T)), skip diagonal ----
    // C/D layout: VGPR e, lane: M = e + 8*half, N = lr.
    float inv_r[8];
    {
        const int rb0 = rowBase + wave * 16 + half * 8;
        #pragma unroll
        for (int e = 0; e < 8; ++e) inv_r[e] = inv_norm[rb0 + e];
    }

    float local = 0.0f;
    #pragma unroll
    for (int n = 0; n < 8; ++n) {
        const int gc = colBase + n * 16 + lr;
        const float ic = inv_norm[gc];
        #pragma unroll
        for (int e = 0; e < 8; ++e) {
            const int gr = rowBase + wave * 16 + half * 8 + e;
            if (gr != gc) {
                float x = acc[n][e] * inv_r[e] * ic - TEMP;
                local += log1pf(expf(x));   // x in [-1.2, 0.8]: safe
            }
        }
    }

    red[t] = local;
    __syncthreads();
    #pragma unroll
    for (int off = 128; off >= 1; off >>= 1) {
        if (t < off) red[t] += red[t + off];
        __syncthreads();
    }
    if (t == 0) {
        const float w = (bi == bj) ? 1.0f : 2.0f;   // symmetry weighting
        atomicAdd(neg_accum, red[0] * w);
    }
}

// ---------------------------------------------------------------------------
// Kernel 3: combine accumulators into the scalar loss.
// ---------------------------------------------------------------------------
__global__ void finalize_kernel(const float* __restrict__ ws, float* __restrict__ out) {
    const double n  = (double)NROWS;
    double pos = (double)ws[0] / n;
    double neg = (double)ws[1] / (n * (n - 1.0));
    out[0] = (float)(0.5 * pos + 0.5 * neg);
}

// ---------------------------------------------------------------------------
extern "C" void kernel_launch(void* const* d_in, const int* in_sizes, int n_in,
                              void* d_out, int out_size, void* d_ws, size_t ws_size,
                              hipStream_t stream) {
    (void)in_sizes; (void)n_in; (void)out_size; (void)ws_size;
    const float* emb = (const float*)d_in[0];
    float* wsf  = (float*)d_ws;
    float* pos  = wsf;       // ws[0]
    float* neg  = wsf + 1;   // ws[1]
    float* invn = wsf + 2;   // ws[2 .. 2+8192)

    zero_accum_kernel<<<1, 32, 0, stream>>>(wsf);
    rownorm_pos_kernel<<<NROWS / 8, 256, 0, stream>>>(emb, invn, pos);
    dim3 grid(NROWS / TILE, NROWS / TILE);
    sim_tile_kernel<<<grid, 256, 0, stream>>>(emb, invn, neg);
    finalize_kernel<<<1, 1, 0, stream>>>(wsf, (float*)d_out);
}